// BGATBlock_72267119722741
// MI455X (gfx1250) — compile-verified
//
#include <hip/hip_runtime.h>
#include <math.h>

#define B_   16
#define M_   128
#define N_   256
#define UD   8
#define AD   2
#define H_   4
#define HD_  32
#define HID  128
#define NEG  0.2f
#define WB_  2.0f
#define DMIN 0.01f
#define HZ_  10.0f
#define PMAX 1.0f
#define SCALE_ 0.17677669529663687f   /* 1/sqrt(32) */

typedef float v2f __attribute__((ext_vector_type(2)));
typedef float v8f __attribute__((ext_vector_type(8)));

__device__ __forceinline__ float leaky(float x){ return x >= 0.f ? x : NEG * x; }
__device__ __forceinline__ float wsum32(float v){
#pragma unroll
  for (int o = 16; o > 0; o >>= 1) v += __shfl_xor(v, o, 32);
  return v;
}
__device__ __forceinline__ float nan2num(float x, float nv, float pv, float ngv){
  if (isnan(x)) return nv;
  if (isinf(x)) return x > 0.f ? pv : ngv;
  return x;
}

// ---------------- K0: per-head rank-1 LN constants for the edge projection ----
__global__ void k0_econsts(const float* __restrict__ We, const float* __restrict__ leg,
                           const float* __restrict__ leb, float* __restrict__ cg,
                           float* __restrict__ be, float* __restrict__ vw){
  int tid = threadIdx.x;            // 128 threads, one wave per head
  int h = tid >> 5, d = tid & 31;
  float w  = We[h * HD_ + d];       // EDGE_DIM == 1
  float mw = wsum32(w) * (1.f / 32.f);
  float c  = w - mw;
  float vv = wsum32(c * c) * (1.f / 32.f);
  cg[h * HD_ + d] = c * leg[h * HD_ + d];
  be[h * HD_ + d] = leb[h * HD_ + d];
  if (d == 0) vw[h] = vv;
}

// ---------------- K1/K2: user / antenna projections, LN over hd=32 (one wave) --
__global__ void k1_projU(const float* __restrict__ user, const float* __restrict__ Wu,
                         const float* __restrict__ g, const float* __restrict__ bb,
                         float* __restrict__ U){
  int row = blockIdx.x * 4 + (threadIdx.x >> 5);   // row over (h,b,m)
  int d   = threadIdx.x & 31;
  int h = row / (B_ * M_); int rem = row % (B_ * M_);
  int b = rem / M_, m = rem % M_;
  float v = 0.f;
#pragma unroll
  for (int i = 0; i < UD; ++i)
    v += user[(b * M_ + m) * UD + i] * Wu[(h * HD_ + d) * UD + i];
  float mean = wsum32(v) * (1.f / 32.f);
  float c = v - mean;
  float var = wsum32(c * c) * (1.f / 32.f);
  float o = c * rsqrtf(var + 1e-5f) * g[h * HD_ + d] + bb[h * HD_ + d];
  U[(size_t)row * HD_ + d] = leaky(o);
}

__global__ void k2_projA(const float* __restrict__ ant, const float* __restrict__ Wa,
                         const float* __restrict__ g, const float* __restrict__ bb,
                         float* __restrict__ A){
  int row = blockIdx.x * 4 + (threadIdx.x >> 5);   // row over (h,b,n)
  int d   = threadIdx.x & 31;
  int h = row / (B_ * N_); int rem = row % (B_ * N_);
  int b = rem / N_, n = rem % N_;
  float v = 0.f;
#pragma unroll
  for (int i = 0; i < AD; ++i)
    v += ant[(b * N_ + n) * AD + i] * Wa[(h * HD_ + d) * AD + i];
  float mean = wsum32(v) * (1.f / 32.f);
  float c = v - mean;
  float var = wsum32(c * c) * (1.f / 32.f);
  float o = c * rsqrtf(var + 1e-5f) * g[h * HD_ + d] + bb[h * HD_ + d];
  A[(size_t)row * HD_ + d] = leaky(o);
}

// ---------------- K3: fused scores (closed-form edge-LN) + softmax over N -----
__global__ void k3_scores(const float* __restrict__ U, const float* __restrict__ A,
                          const float* __restrict__ edge, const float* __restrict__ cg,
                          const float* __restrict__ be, const float* __restrict__ vw,
                          const float* __restrict__ av, float* __restrict__ alpha){
  int id = blockIdx.x;                       // (h,b,m)
  int h = id / (B_ * M_); int rem = id % (B_ * M_);
  int b = rem / M_, m = rem % M_;
  int n = threadIdx.x;                       // 256 threads = N
  __shared__ float Urow[HD_], cgs[HD_], bes[HD_], avs[HD_], red[N_];
  if (n < HD_) {
    Urow[n] = U[((size_t)((h * B_ + b) * M_ + m)) * HD_ + n];
    cgs[n]  = cg[h * HD_ + n];
    bes[n]  = be[h * HD_ + n];
    avs[n]  = av[h * HD_ + n];
  }
  __syncthreads();
  float vwh = vw[h];
  float e = edge[(size_t)(b * M_ + m) * N_ + n];
  float s = e * rsqrtf(e * e * vwh + 1e-5f);
  const float4* Ar = (const float4*)(A + ((size_t)((h * B_ + b) * N_ + n)) * HD_);
  float score = 0.f;
#pragma unroll
  for (int q = 0; q < 8; ++q) {
    float4 a4 = Ar[q];
    float aa[4] = {a4.x, a4.y, a4.z, a4.w};
#pragma unroll
    for (int j = 0; j < 4; ++j) {
      int d = q * 4 + j;
      float E = leaky(s * cgs[d] + bes[d]);
      score += avs[d] * leaky(Urow[d] + aa[j] + E);
    }
  }
  score *= SCALE_;
  // block softmax over N=256 (max-subtract, like the reference)
  red[n] = score; __syncthreads();
  for (int st = 128; st > 0; st >>= 1) { if (n < st) red[n] = fmaxf(red[n], red[n + st]); __syncthreads(); }
  float mx = red[0]; __syncthreads();
  float ex = expf(score - mx);
  red[n] = ex; __syncthreads();
  for (int st = 128; st > 0; st >>= 1) { if (n < st) red[n] += red[n + st]; __syncthreads(); }
  float sm = red[0];
  alpha[((size_t)((h * B_ + b) * M_ + m)) * N_ + n] = ex / sm;
}

// ---------------- K4: ant_out = alpha^T @ U via V_WMMA_F32_16X16X4_F32 --------
// per (h,b): C[256,32] = alpha^T(256x128) * U(128x32); residual+ReLU fused.
__global__ void k4_antout_wmma(const float* __restrict__ alpha, const float* __restrict__ U,
                               const float* __restrict__ ant, const float* __restrict__ Wra,
                               const float* __restrict__ bra, float* __restrict__ apre){
  int hb = blockIdx.x;                 // h*B + b
  int h = hb / B_, b = hb % B_;
  __shared__ float Us[M_ * HD_];       // 16 KB: U tile for this (h,b)
  int tid = threadIdx.x;
  const float4* Ug = (const float4*)(U + (size_t)hb * M_ * HD_);
  for (int i = tid; i < M_ * HD_ / 4; i += 256) ((float4*)Us)[i] = Ug[i];
  __syncthreads();
  int wave = tid >> 5, lane = tid & 31;
  int hs = lane >> 4, lr = lane & 15;  // lane-half select, row/col-in-half
  const float* Ab = alpha + (size_t)hb * M_ * N_;
  for (int t = wave; t < 16; t += 8) { // 16 n-tiles
    v8f acc0 = {}; v8f acc1 = {};
    int nrow = t * 16 + lr;
    for (int k = 0; k < M_; k += 4) {
      int ka = k + 2 * hs;
      v2f a;  a.x  = Ab[(size_t)(ka    ) * N_ + nrow];
              a.y  = Ab[(size_t)(ka + 1) * N_ + nrow];
      v2f b0; b0.x = Us[(ka) * HD_ + lr];       b0.y = Us[(ka + 1) * HD_ + lr];
      v2f b1; b1.x = Us[(ka) * HD_ + 16 + lr];  b1.y = Us[(ka + 1) * HD_ + 16 + lr];
      acc0 = __builtin_amdgcn_wmma_f32_16x16x4_f32(false, a, false, b0, (short)0, acc0, false, false);
      acc1 = __builtin_amdgcn_wmma_f32_16x16x4_f32(false, a, false, b1, (short)0, acc1, false, false);
    }
#pragma unroll
    for (int r = 0; r < 8; ++r) {
      int n = t * 16 + r + 8 * hs;
      float a0 = ant[(b * N_ + n) * AD + 0], a1 = ant[(b * N_ + n) * AD + 1];
#pragma unroll
      for (int dt = 0; dt < 2; ++dt) {
        int c = h * HD_ + dt * 16 + lr;
        float res = leaky(a0 * Wra[c * AD + 0] + a1 * Wra[c * AD + 1] + bra[c]);
        float v = (dt == 0 ? acc0[r] : acc1[r]) + res;
        apre[((size_t)(b * N_ + n)) * HID + c] = fmaxf(v, 0.f);
      }
    }
  }
}

// ---------------- K5: LayerNorm over HIDDEN=128 per (b,n) ---------------------
__global__ void k5_ln(const float* __restrict__ apre, const float* __restrict__ g,
                      const float* __restrict__ bb, float* __restrict__ aat){
  int row = blockIdx.x, c = threadIdx.x;   // 128 threads
  __shared__ float red[HID];
  float v = apre[(size_t)row * HID + c];
  red[c] = v; __syncthreads();
  for (int st = 64; st > 0; st >>= 1) { if (c < st) red[c] += red[c + st]; __syncthreads(); }
  float mean = red[0] * (1.f / HID); __syncthreads();
  float d = v - mean;
  red[c] = d * d; __syncthreads();
  for (int st = 64; st > 0; st >>= 1) { if (c < st) red[c] += red[c + st]; __syncthreads(); }
  float var = red[0] * (1.f / HID);
  aat[(size_t)row * HID + c] = d * rsqrtf(var + 1e-5f) * g[c] + bb[c];
}

// ---------------- K6: h1 = relu(a_attn @ Wm1^T + bm1) via f32 WMMA ------------
// M=4096 rows, N=64 cols, K=128. Wm1 staged in LDS.
__global__ void k6_mlp1_wmma(const float* __restrict__ aat, const float* __restrict__ Wm1,
                             const float* __restrict__ bm1, float* __restrict__ h1){
  __shared__ float Wm1s[64 * HID];     // 32 KB
  int tid = threadIdx.x;
  for (int i = tid; i < 64 * HID / 4; i += 256) ((float4*)Wm1s)[i] = ((const float4*)Wm1)[i];
  __syncthreads();
  int wave = tid >> 5, lane = tid & 31;
  int hs = lane >> 4, lr = lane & 15;
  int t = blockIdx.x * 8 + wave;       // 1024 tiles total
  int mt = t >> 2, nt = t & 3;
  v8f acc = {};
  int arow = mt * 16 + lr;
  int col  = nt * 16 + lr;
  for (int k = 0; k < HID; k += 4) {
    int ka = k + 2 * hs;
    v2f a;  a.x = aat[(size_t)arow * HID + ka];  a.y = aat[(size_t)arow * HID + ka + 1];
    v2f bmat; bmat.x = Wm1s[col * HID + ka];     bmat.y = Wm1s[col * HID + ka + 1];
    acc = __builtin_amdgcn_wmma_f32_16x16x4_f32(false, a, false, bmat, (short)0, acc, false, false);
  }
#pragma unroll
  for (int r = 0; r < 8; ++r) {
    int row = mt * 16 + r + 8 * hs;
    h1[(size_t)row * 64 + col] = fmaxf(acc[r] + bm1[col], 0.f);
  }
}

// ---------------- K7: per-batch readouts, scan, delta/power paths -------------
__global__ void k7_readout(const float* __restrict__ h1, const float* __restrict__ Wm2,
                           const float* __restrict__ bm2, const float* __restrict__ Wd1,
                           const float* __restrict__ bd1, const float* __restrict__ Wd2,
                           const float* __restrict__ bd2, const float* __restrict__ Wp1,
                           const float* __restrict__ bp1, const float* __restrict__ Wp2,
                           const float* __restrict__ bp2, float* __restrict__ out_af,
                           float* __restrict__ out_pos, float* __restrict__ xbuf){
  int b = blockIdx.x, tid = threadIdx.x;   // 512 threads
  __shared__ float da[N_], pa[N_], hid[512], red[512], scn[N_];
  // ant2d = clip(relu(h1 @ Wm2^T + bm2), -200, 200)
  if (tid < N_) {
    const float* hr = h1 + (size_t)(b * N_ + tid) * 64;
    float d0 = bm2[0], d1 = bm2[1];
    for (int j = 0; j < 64; ++j) { float hv = hr[j]; d0 += hv * Wm2[j]; d1 += hv * Wm2[64 + j]; }
    da[tid] = fminf(fmaxf(fmaxf(d0, 0.f), -200.f), 200.f);
    pa[tid] = fminf(fmaxf(fmaxf(d1, 0.f), -200.f), 200.f);
  }
  __syncthreads();
  // delta readout: 256 -> 512 -> 256
  { float acc = bd1[tid];
    for (int n = 0; n < N_; ++n) acc += da[n] * Wd1[tid * N_ + n];
    hid[tid] = fmaxf(acc, 0.f); }
  __syncthreads();
  float rd = 0.f;
  if (tid < N_) {
    float acc = bd2[tid];
    for (int j = 0; j < 512; ++j) acc += hid[j] * Wd2[tid * 512 + j];
    rd = fminf(fmaxf(acc, 0.f), 100.f);          // clip(rd,0,100); relu is no-op after
  }
  __syncthreads();
  if (tid < N_) da[tid] = rd;                    // reuse as delta activations
  __syncthreads();
  // power readout: 256 -> 512 -> 256
  { float acc = bp1[tid];
    for (int n = 0; n < N_; ++n) acc += pa[n] * Wp1[tid * N_ + n];
    hid[tid] = fmaxf(acc, 0.f); }
  __syncthreads();
  float pav = 0.f;
  if (tid < N_) {
    float acc = bp2[tid];
    for (int j = 0; j < 512; ++j) acc += hid[j] * Wp2[tid * 512 + j];
    float rp = nan2num(acc, 0.f, 1.f, 0.f);
    pav = fminf(fmaxf(nan2num(fmaxf(rp, 0.f), 0.f, 1.f, 0.f), 0.f), 1.f);
  }
  __syncthreads();
  if (tid < N_) pa[tid] = pav;
  __syncthreads();
  // sums
  red[tid] = (tid < N_) ? da[tid] : 0.f; __syncthreads();
  for (int st = 256; st > 0; st >>= 1) { if (tid < st) red[tid] += red[tid + st]; __syncthreads(); }
  float sd = red[0]; __syncthreads();
  red[tid] = (tid < N_) ? pa[tid] : 0.f; __syncthreads();
  for (int st = 256; st > 0; st >>= 1) { if (tid < st) red[tid] += red[tid + st]; __syncthreads(); }
  float sp = red[0]; __syncthreads();
  // delta path
  const float Bmax = 2.f * WB_ - (N_ - 1) * DMIN;   // 1.45
  float sdelta = 0.f;
  if (tid < N_) sdelta = fminf(fmaxf((Bmax / fmaxf(Bmax, sd)) * da[tid], 0.f), 100.f);
  if (tid < N_) scn[tid] = sdelta;
  __syncthreads();
  for (int off = 1; off < N_; off <<= 1) {          // inclusive scan (cumsum)
    float tv = 0.f;
    if (tid < N_ && tid >= off) tv = scn[tid - off];
    __syncthreads();
    if (tid < N_) scn[tid] += tv;
    __syncthreads();
  }
  // power path
  float spn = nan2num(sp, 1e-3f, 1e-3f, -1e-3f);
  float denom = fmaxf(spn, 1e-3f);
  float sfp = nan2num(PMAX / denom, 1.f, 1.f, 0.f);
  if (tid < N_) {
    float spow = fminf(fmaxf(nan2num(sfp * pa[tid], 0.f, 1.f, 0.f), 0.f), 1.f);
    float x = scn[tid] + tid * DMIN - (tid + 1.f) * WB_;
    int gi = b * N_ + tid;
    out_af[gi * 2 + 0] = sdelta;
    out_af[gi * 2 + 1] = spow;
    out_pos[gi * 3 + 0] = x;
    out_pos[gi * 3 + 1] = 0.f;
    out_pos[gi * 3 + 2] = HZ_;
    xbuf[gi] = x;
  }
}

// ---------------- K8: edge_feats_new = ||user_xy - pos_xy|| -------------------
__global__ void k8_edge(const float* __restrict__ user, const float* __restrict__ xbuf,
                        float* __restrict__ out_edge){
  int idx = blockIdx.x * 256 + threadIdx.x;
  if (idx >= B_ * M_ * N_) return;
  int b = idx / (M_ * N_); int rem = idx % (M_ * N_);
  int m = rem / N_, n = rem % N_;
  float dx = user[(b * M_ + m) * UD + 0] - xbuf[b * N_ + n];
  float dy = user[(b * M_ + m) * UD + 1];   // positions y == 0
  out_edge[idx] = sqrtf(dx * dx + dy * dy);
}

extern "C" void kernel_launch(void* const* d_in, const int* in_sizes, int n_in,
                              void* d_out, int out_size, void* d_ws, size_t ws_size,
                              hipStream_t stream) {
  const float* user = (const float*)d_in[0];
  const float* ant  = (const float*)d_in[1];
  const float* edge = (const float*)d_in[2];
  const float* Wu   = (const float*)d_in[3];
  const float* lug  = (const float*)d_in[4];
  const float* lub  = (const float*)d_in[5];
  const float* Wa   = (const float*)d_in[6];
  const float* lag  = (const float*)d_in[7];
  const float* lab  = (const float*)d_in[8];
  const float* We   = (const float*)d_in[9];
  const float* leg  = (const float*)d_in[10];
  const float* leb  = (const float*)d_in[11];
  const float* av   = (const float*)d_in[12];
  const float* Wra  = (const float*)d_in[13];
  const float* bra  = (const float*)d_in[14];
  const float* lng  = (const float*)d_in[15];
  const float* lnb  = (const float*)d_in[16];
  const float* Wm1  = (const float*)d_in[17];
  const float* bm1  = (const float*)d_in[18];
  const float* Wm2  = (const float*)d_in[19];
  const float* bm2  = (const float*)d_in[20];
  const float* Wd1  = (const float*)d_in[21];
  const float* bd1  = (const float*)d_in[22];
  const float* Wd2  = (const float*)d_in[23];
  const float* bd2  = (const float*)d_in[24];
  const float* Wp1  = (const float*)d_in[25];
  const float* bp1  = (const float*)d_in[26];
  const float* Wp2  = (const float*)d_in[27];
  const float* bp2  = (const float*)d_in[28];

  float* ws    = (float*)d_ws;
  float* cg    = ws;                         // 128
  float* be    = cg + 128;                   // 128
  float* vw    = be + 128;                   // 4
  float* Ubuf  = vw + 4;                     // H*B*M*HD   = 262144
  float* Abuf  = Ubuf + H_ * B_ * M_ * HD_;  // H*B*N*HD   = 524288
  float* alpha = Abuf + H_ * B_ * N_ * HD_;  // H*B*M*N    = 2097152
  float* apre  = alpha + (size_t)H_ * B_ * M_ * N_;  // B*N*HID = 524288
  float* aat   = apre + (size_t)B_ * N_ * HID;       // B*N*HID = 524288
  float* h1    = aat + (size_t)B_ * N_ * HID;        // B*N*64  = 262144
  float* xbuf  = h1 + (size_t)B_ * N_ * 64;          // B*N     = 4096

  float* out      = (float*)d_out;
  float* out_af   = out;                       // [B,N,2]
  float* out_edge = out + B_ * N_ * 2;         // [B,M,N,1]
  float* out_pos  = out_edge + B_ * M_ * N_;   // [B,N,3]

  k0_econsts<<<1, 128, 0, stream>>>(We, leg, leb, cg, be, vw);
  k1_projU<<<(H_ * B_ * M_) / 4, 128, 0, stream>>>(user, Wu, lug, lub, Ubuf);
  k2_projA<<<(H_ * B_ * N_) / 4, 128, 0, stream>>>(ant, Wa, lag, lab, Abuf);
  k3_scores<<<H_ * B_ * M_, 256, 0, stream>>>(Ubuf, Abuf, edge, cg, be, vw, av, alpha);
  k4_antout_wmma<<<H_ * B_, 256, 0, stream>>>(alpha, Ubuf, ant, Wra, bra, apre);
  k5_ln<<<B_ * N_, 128, 0, stream>>>(apre, lng, lnb, aat);
  k6_mlp1_wmma<<<128, 256, 0, stream>>>(aat, Wm1, bm1, h1);
  k7_readout<<<B_, 512, 0, stream>>>(h1, Wm2, bm2, Wd1, bd1, Wd2, bd2,
                                     Wp1, bp1, Wp2, bp2, out_af, out_pos, xbuf);
  k8_edge<<<(B_ * M_ * N_ + 255) / 256, 256, 0, stream>>>(user, xbuf, out_edge);
}